// GIN_30520037605492
// MI455X (gfx1250) — compile-verified
//
#include <hip/hip_runtime.h>

typedef __attribute__((ext_vector_type(2))) float v2f;
typedef __attribute__((ext_vector_type(8))) float v8f;

#define NN 100000
#define EE 1600000
#define DD 128
#define BN_EPS 1e-5f

// ---------------------------------------------------------------------------
// small utility kernels
// ---------------------------------------------------------------------------
__global__ __launch_bounds__(256) void zero_kernel(float* p, int n) {
  int i = blockIdx.x * 256 + threadIdx.x;
  if (i < n) p[i] = 0.f;
}

__global__ __launch_bounds__(256) void copy_kernel(const float4* __restrict__ x,
                                                   float4* __restrict__ o, int n4) {
  int i = blockIdx.x * 256 + threadIdx.x;
  if (i < n4) o[i] = x[i];
}

// one wave per edge: lane handles 4 contiguous floats of the 128-wide row.
// unsafeAtomicAdd -> native global_atomic_add_f32 (no CAS loop).
__global__ __launch_bounds__(256) void scatter_kernel(const float* __restrict__ h,
                                                      const int* __restrict__ ei,
                                                      float* __restrict__ out) {
  int e = blockIdx.x * 8 + (threadIdx.x >> 5);
  if (e >= EE) return;
  int lane = threadIdx.x & 31;
  int src = ei[e];
  int dst = ei[EE + e];
  const float4 v = *(const float4*)(h + (size_t)src * DD + lane * 4);
  float* o = out + (size_t)dst * DD + lane * 4;
  unsafeAtomicAdd(o + 0, v.x);
  unsafeAtomicAdd(o + 1, v.y);
  unsafeAtomicAdd(o + 2, v.z);
  unsafeAtomicAdd(o + 3, v.w);
}

// fold BN statistics + gamma/beta into per-column scale/shift (128 threads)
__global__ void finalize_kernel(const float* __restrict__ sum, const float* __restrict__ sq,
                                const float* __restrict__ g, const float* __restrict__ be,
                                float* __restrict__ scale, float* __restrict__ shift) {
  int j = threadIdx.x;
  const float invN = 1.f / (float)NN;
  float m = sum[j] * invN;
  float v = fmaf(-m, m, sq[j] * invN);   // biased variance
  float s = g[j] * rsqrtf(v + BN_EPS);
  scale[j] = s;
  shift[j] = fmaf(-m, s, be[j]);
}

// h = relu(t*scale+shift); duplicate into out1 (scatter base). out1 may alias t.
__global__ __launch_bounds__(256) void bnrelu_dup_kernel(const float* __restrict__ t,
                                                         const float* __restrict__ scale,
                                                         const float* __restrict__ shift,
                                                         float* __restrict__ h,
                                                         float* __restrict__ out1, int n4) {
  int i = blockIdx.x * 256 + threadIdx.x;
  if (i >= n4) return;
  int c = (i & 31) * 4;
  float4 v = ((const float4*)t)[i];
  v.x = fmaxf(fmaf(v.x, scale[c + 0], shift[c + 0]), 0.f);
  v.y = fmaxf(fmaf(v.y, scale[c + 1], shift[c + 1]), 0.f);
  v.z = fmaxf(fmaf(v.z, scale[c + 2], shift[c + 2]), 0.f);
  v.w = fmaxf(fmaf(v.w, scale[c + 3], shift[c + 3]), 0.f);
  ((float4*)h)[i] = v;
  ((float4*)out1)[i] = v;
}

// ---------------------------------------------------------------------------
// fused GEMM: C = op(A) @ W + bias, op = identity or relu(a*scale+shift)
// optionally accumulates per-column sum / sum-of-squares (for next BN).
// 256 threads = 8 waves, 128-row M-tile. fp32 WMMA (V_WMMA_F32_16X16X4_F32).
// LDS manually laid out: W (k-pair-interleaved float2) at offset 0 so the hot
// loop's ds_load_b64 offsets are 16-bit immediates; A tile in the upper 64KB.
// ---------------------------------------------------------------------------
template <bool AFFINE, bool STATS>
__global__ __launch_bounds__(256) void gemm_kernel(
    const float* __restrict__ A, const float* __restrict__ W,
    const float* __restrict__ bias, const float* __restrict__ scale,
    const float* __restrict__ shift, float* __restrict__ C,
    float* __restrict__ gsum, float* __restrict__ gsq) {
  __shared__ __align__(16) float lds[2 * DD * DD];  // [0,64K): W2, [64K,128K): A
  __shared__ float sSum[DD];
  __shared__ float sSq[DD];
  __shared__ float sBias[DD];

  const int tid = threadIdx.x;
  const int m0 = blockIdx.x * 128;
  float* ldsA = lds + DD * DD;

  // stage W pair-interleaved: lds2[kp*128+n] = {W[2kp][n], W[2kp+1][n]}
  {
    float2* lds2 = (float2*)lds;
    for (int i = tid; i < (DD / 2) * DD; i += 256) {
      const int kp = i >> 7;
      const int n = i & 127;
      float2 w2;
      w2.x = W[(size_t)(2 * kp) * DD + n];
      w2.y = W[(size_t)(2 * kp + 1) * DD + n];
      lds2[i] = w2;  // ds_store_b64, lanes hit distinct banks
    }
  }
  // stage A tile, applying BN affine + relu on the fly; OOB rows -> 0
  for (int i = tid; i < DD * DD / 4; i += 256) {
    int row = i >> 5;
    int c = (i & 31) * 4;
    float4 v = make_float4(0.f, 0.f, 0.f, 0.f);
    if (m0 + row < NN) {
      v = *(const float4*)(A + (size_t)(m0 + row) * DD + c);
      if (AFFINE) {
        v.x = fmaxf(fmaf(v.x, scale[c + 0], shift[c + 0]), 0.f);
        v.y = fmaxf(fmaf(v.y, scale[c + 1], shift[c + 1]), 0.f);
        v.z = fmaxf(fmaf(v.z, scale[c + 2], shift[c + 2]), 0.f);
        v.w = fmaxf(fmaf(v.w, scale[c + 3], shift[c + 3]), 0.f);
      }
    }
    *(float4*)(ldsA + row * DD + c) = v;
  }
  if (tid < DD) {
    sBias[tid] = bias[tid];
    if (STATS) { sSum[tid] = 0.f; sSq[tid] = 0.f; }
  }
  __syncthreads();

  const int wave = tid >> 5;
  const int lane = tid & 31;
  const int col = lane & 15;
  const int half = lane >> 4;
  const int koff = half << 1;            // A/B f32 fragment: K offset 0 or 2
  const int mrow = (wave << 4) + col;    // A-fragment row within tile
  const int mhalf = half << 3;           // C rows: i or i+8 per half-wave

  // preload this wave's 16x128 A slice as 32 K-fragments (one-time ds_load_b64s)
  v2f afrag[32];
  const float* aptr = ldsA + mrow * DD + koff;
#pragma unroll
  for (int k0 = 0; k0 < 32; ++k0)
    afrag[k0] = *(const v2f*)(aptr + (k0 << 2));

  // B fragment base: pair (2*k0 + half), column n -> float2 index (2k0+half)*128+n
  const v2f* bbase = ((const v2f*)lds) + (half << 7);

  for (int nt = 0; nt < 8; nt += 2) {
    const int n0 = (nt << 4) + col;
    const v2f* bp0 = bbase + n0;
    const v2f* bp1 = bp0 + 16;
    v8f acc0 = {}, acc1 = {};
#pragma unroll
    for (int k0 = 0; k0 < 32; ++k0) {
      const v2f b0 = bp0[k0 << 8];       // byte offset k0*2048: DS immediate
      const v2f b1 = bp1[k0 << 8];
      acc0 = __builtin_amdgcn_wmma_f32_16x16x4_f32(false, afrag[k0], false, b0,
                                                   (short)0, acc0, false, false);
      acc1 = __builtin_amdgcn_wmma_f32_16x16x4_f32(false, afrag[k0], false, b1,
                                                   (short)0, acc1, false, false);
    }
#pragma unroll
    for (int t = 0; t < 2; ++t) {
      const int n = n0 + (t << 4);
      const v8f accv = t ? acc1 : acc0;
      const float bv = sBias[n];
      float psum = 0.f, psq = 0.f;
#pragma unroll
      for (int i = 0; i < 8; ++i) {
        const int m = m0 + (wave << 4) + mhalf + i;
        const float val = accv[i] + bv;
        if (m < NN) {
          C[(size_t)m * DD + n] = val;
          if (STATS) { psum += val; psq += val * val; }
        }
      }
      if (STATS) {
        atomicAdd(&sSum[n], psum);       // ds_add_f32
        atomicAdd(&sSq[n], psq);
      }
    }
  }
  if (STATS) {
    __syncthreads();
    if (tid < DD) {
      unsafeAtomicAdd(&gsum[tid], sSum[tid]);
      unsafeAtomicAdd(&gsq[tid], sSq[tid]);
    }
  }
}

// ---------------------------------------------------------------------------
extern "C" void kernel_launch(void* const* d_in, const int* in_sizes, int n_in,
                              void* d_out, int out_size, void* d_ws, size_t ws_size,
                              hipStream_t stream) {
  const float* x   = (const float*)d_in[0];
  const int*   ei  = (const int*)d_in[1];
  const float* W1a = (const float*)d_in[2];
  const float* b1a = (const float*)d_in[3];
  const float* g1a = (const float*)d_in[4];
  const float* be1a= (const float*)d_in[5];
  const float* W1b = (const float*)d_in[6];
  const float* b1b = (const float*)d_in[7];
  const float* g1c = (const float*)d_in[8];
  const float* be1c= (const float*)d_in[9];
  const float* W2a = (const float*)d_in[10];
  const float* b2a = (const float*)d_in[11];
  const float* g2a = (const float*)d_in[12];
  const float* be2a= (const float*)d_in[13];
  const float* W2b = (const float*)d_in[14];
  const float* b2b = (const float*)d_in[15];
  float* out = (float*)d_out;

  // workspace layout: two N x D fp32 buffers + stats + scale/shift
  float* bufA = (float*)d_ws;                         // N*D
  float* bufB = bufA + (size_t)NN * DD;               // N*D
  float* st   = bufB + (size_t)NN * DD;               // 6*128 sums/sqs
  float* s1 = st + 0 * DD, *q1 = st + 1 * DD;
  float* s2 = st + 2 * DD, *q2 = st + 3 * DD;
  float* s3 = st + 4 * DD, *q3 = st + 5 * DD;
  float* sc = st + 6 * DD;                            // 6*128 scale/shift
  float* sc1 = sc + 0 * DD, *sh1 = sc + 1 * DD;
  float* sc2 = sc + 2 * DD, *sh2 = sc + 3 * DD;
  float* sc3 = sc + 4 * DD, *sh3 = sc + 5 * DD;

  const int n4 = NN * (DD / 4);                 // float4 count per N x D buffer
  const int gemmBlocks = (NN + 127) / 128;      // 782
  const int edgeBlocks = (EE + 7) / 8;          // 200000

  // clear statistics accumulators (ws is poisoned between runs)
  zero_kernel<<<(6 * DD + 255) / 256, 256, 0, stream>>>(st, 6 * DD);

  // ---- layer 0: out0 = x + agg(x)  (bufA) ----
  copy_kernel<<<(n4 + 255) / 256, 256, 0, stream>>>((const float4*)x, (float4*)bufA, n4);
  scatter_kernel<<<edgeBlocks, 256, 0, stream>>>(x, ei, bufA);

  // t1 = out0 @ W1a + b1a (bufB), stats -> s1/q1
  gemm_kernel<false, true><<<gemmBlocks, 256, 0, stream>>>(bufA, W1a, b1a, nullptr,
                                                           nullptr, bufB, s1, q1);
  finalize_kernel<<<1, DD, 0, stream>>>(s1, q1, g1a, be1a, sc1, sh1);

  // t2 = relu(bn(t1)) @ W1b + b1b (bufA), stats -> s2/q2
  gemm_kernel<true, true><<<gemmBlocks, 256, 0, stream>>>(bufB, W1b, b1b, sc1, sh1,
                                                          bufA, s2, q2);
  finalize_kernel<<<1, DD, 0, stream>>>(s2, q2, g1c, be1c, sc2, sh2);

  // h = relu(bn(t2)) (bufB); out1 initialized to h (bufA, aliases t2 safely)
  bnrelu_dup_kernel<<<(n4 + 255) / 256, 256, 0, stream>>>(bufA, sc2, sh2, bufB, bufA, n4);
  // out1 += agg(h)
  scatter_kernel<<<edgeBlocks, 256, 0, stream>>>(bufB, ei, bufA);

  // t3 = out1 @ W2a + b2a (bufB), stats -> s3/q3
  gemm_kernel<false, true><<<gemmBlocks, 256, 0, stream>>>(bufA, W2a, b2a, nullptr,
                                                           nullptr, bufB, s3, q3);
  finalize_kernel<<<1, DD, 0, stream>>>(s3, q3, g2a, be2a, sc3, sh3);

  // out = relu(bn(t3)) @ W2b + b2b
  gemm_kernel<true, false><<<gemmBlocks, 256, 0, stream>>>(bufB, W2b, b2b, sc3, sh3,
                                                           out, nullptr, nullptr);
}